// GeospatialGraphNN_18476949307919
// MI455X (gfx1250) — compile-verified
//
#include <hip/hip_runtime.h>
#include <hip/hip_bf16.h>

#define HID    512
#define NLAY   3
#define KNN    8
#define NB     4
#define NN     4096
#define REVCAP 64

typedef float v2f __attribute__((ext_vector_type(2)));
typedef float v8f __attribute__((ext_vector_type(8)));

// ---------------------------------------------------------------------------
// 1) brute-force kNN (k=8) over 2-D coords, per batch. One thread per node,
//    coord tiles staged through LDS. Keeps top-9 (incl. self), drops self.
// ---------------------------------------------------------------------------
__global__ void __launch_bounds__(256) knn_kernel(const float2* __restrict__ coords,
                                                  int* __restrict__ knn)
{
    const int gid = blockIdx.x * 256 + threadIdx.x;      // b*NN + n
    const int b   = gid / NN;
    const int n   = gid - b * NN;
    const float2 me = coords[gid];

    float bd[KNN + 1];
    int   bi[KNN + 1];
#pragma unroll
    for (int i = 0; i <= KNN; ++i) { bd[i] = 3.4e38f; bi[i] = -1; }

    __shared__ float2 tile[256];
    const float2* cb = coords + (size_t)b * NN;          // whole block is same b
    for (int base = 0; base < NN; base += 256) {
        __syncthreads();
        tile[threadIdx.x] = cb[base + threadIdx.x];
        __syncthreads();
        for (int t = 0; t < 256; ++t) {
            const float dx = me.x - tile[t].x;
            const float dy = me.y - tile[t].y;
            const float d  = dx * dx + dy * dy;
            const int   j  = base + t;
            if (d < bd[KNN] || (d == bd[KNN] && j < bi[KNN])) {
                int p = KNN;
                while (p > 0 && (bd[p - 1] > d ||
                                 (bd[p - 1] == d && bi[p - 1] > j))) {
                    bd[p] = bd[p - 1]; bi[p] = bi[p - 1]; --p;
                }
                bd[p] = d; bi[p] = j;
            }
        }
    }
    int out = 0;
    for (int p = 0; p <= KNN && out < KNN; ++p)
        if (bi[p] != n) knn[(size_t)gid * KNN + (out++)] = bi[p];
}

// ---------------------------------------------------------------------------
// 2) reverse-edge lists (for the symmetric union adjacency)
// ---------------------------------------------------------------------------
__global__ void __launch_bounds__(256) rev_init(int* __restrict__ revcnt)
{
    revcnt[blockIdx.x * 256 + threadIdx.x] = 0;
}

__global__ void __launch_bounds__(256) rev_build(const int* __restrict__ knn,
                                                 int* __restrict__ revcnt,
                                                 int* __restrict__ rev)
{
    const int gid = blockIdx.x * 256 + threadIdx.x;      // edge id: (b*NN+n)*8+s
    const int bn  = gid >> 3;
    const int b   = bn / NN;
    const int n   = bn - b * NN;
    const int j   = knn[gid];                            // target node (in-batch)
    const int slot = atomicAdd(&revcnt[b * NN + j], 1);
    if (slot < REVCAP) rev[(size_t)(b * NN + j) * REVCAP + slot] = n;
}

// ---------------------------------------------------------------------------
// 3) aggregation: agg[n] = sum over (out ∪ rev, dedup'd) neighbors of x[j]
//    One block per node, 128 threads * float4 = 512 features.
// ---------------------------------------------------------------------------
__global__ void __launch_bounds__(128) aggregate(const float* __restrict__ x,
                                                 const int* __restrict__ knn,
                                                 const int* __restrict__ revcnt,
                                                 const int* __restrict__ rev,
                                                 float* __restrict__ agg)
{
    const int bn = blockIdx.x;                 // b*NN + n
    const int b  = bn / NN;

    __shared__ int nbrs[KNN + REVCAP];
    __shared__ int deg_s;
    if (threadIdx.x == 0) {
        int outl[KNN];
        int cnt = 0;
#pragma unroll
        for (int s = 0; s < KNN; ++s) {
            outl[s] = knn[(size_t)bn * KNN + s];
            nbrs[cnt++] = outl[s];
        }
        int rc = revcnt[bn];
        rc = rc < REVCAP ? rc : REVCAP;
        for (int s = 0; s < rc; ++s) {
            const int j = rev[(size_t)bn * REVCAP + s];
            bool dup = false;
#pragma unroll
            for (int t = 0; t < KNN; ++t) dup |= (outl[t] == j);
            if (!dup) nbrs[cnt++] = j;
        }
        deg_s = cnt;
    }
    __syncthreads();
    const int deg = deg_s;
    const int h   = threadIdx.x * 4;

    float ax = 0.f, ay = 0.f, az = 0.f, aw = 0.f;
    for (int s = 0; s < deg; ++s) {
        const float4 v = *(const float4*)&x[((size_t)b * NN + nbrs[s]) * HID + h];
        ax += v.x; ay += v.y; az += v.z; aw += v.w;
    }
    float4 o; o.x = ax; o.y = ay; o.z = az; o.w = aw;
    *(float4*)&agg[(size_t)bn * HID + h] = o;
}

// ---------------------------------------------------------------------------
// 4) W transpose (Wt[n][k] = W[k][n]) so B-matrix WMMA loads are contiguous
// ---------------------------------------------------------------------------
__global__ void __launch_bounds__(256) transposeW(const float* __restrict__ W,
                                                  float* __restrict__ Wt)
{
    __shared__ float t[16][17];
    const int l  = blockIdx.z;
    const int k0 = blockIdx.y * 16;
    const int n0 = blockIdx.x * 16;
    t[threadIdx.y][threadIdx.x] =
        W[(size_t)l * HID * HID + (size_t)(k0 + threadIdx.y) * HID + n0 + threadIdx.x];
    __syncthreads();
    Wt[(size_t)l * HID * HID + (size_t)(n0 + threadIdx.y) * HID + k0 + threadIdx.x] =
        t[threadIdx.x][threadIdx.y];
}

// ---------------------------------------------------------------------------
// 5) GEMM: Y[M x 512] = Agg[M x 512] @ W[512 x 512]   (fp32 WMMA 16x16x4)
//    Block 256 thr = 8 waves laid out 4(M) x 2(N); wave tile = 32 x 64.
//    VGPR layout (ISA 7.12.2, 32-bit A 16x4 / dual for B):
//      lane<16 : K = k..k+1 ; lane>=16 : K = k+2..k+3 ; row/col = lane&15.
// ---------------------------------------------------------------------------
__global__ void __launch_bounds__(256) gemm_wmma(const float* __restrict__ A,
                                                 const float* __restrict__ Wt,
                                                 float* __restrict__ Y)
{
    const int lane = threadIdx.x & 31;
    const int wid  = threadIdx.x >> 5;
    const int wm   = wid & 3;                  // 0..3 : M sub-tile
    const int wn   = wid >> 2;                 // 0..1 : N sub-tile
    const int m0   = blockIdx.x * 128 + wm * 32;
    const int n0   = blockIdx.y * 128 + wn * 64;

    const int lr = lane & 15;                  // row/col within 16-tile
    const int lk = (lane >> 4) * 2;            // K sub-offset (0 or 2)

    const float* a0p = A  + (size_t)(m0 + lr)      * HID + lk;
    const float* a1p = a0p + (size_t)16 * HID;
    const float* b0p = Wt + (size_t)(n0 +  0 + lr) * HID + lk;
    const float* b1p = Wt + (size_t)(n0 + 16 + lr) * HID + lk;
    const float* b2p = Wt + (size_t)(n0 + 32 + lr) * HID + lk;
    const float* b3p = Wt + (size_t)(n0 + 48 + lr) * HID + lk;

    v8f acc[2][4] = {};

#pragma unroll 4
    for (int k = 0; k < HID; k += 4) {
        const v2f a0 = *(const v2f*)(a0p + k);
        const v2f a1 = *(const v2f*)(a1p + k);
        const v2f b0 = *(const v2f*)(b0p + k);
        const v2f b1 = *(const v2f*)(b1p + k);
        const v2f b2 = *(const v2f*)(b2p + k);
        const v2f b3 = *(const v2f*)(b3p + k);
        acc[0][0] = __builtin_amdgcn_wmma_f32_16x16x4_f32(false, a0, false, b0,
                        (short)0, acc[0][0], false, false);
        acc[0][1] = __builtin_amdgcn_wmma_f32_16x16x4_f32(false, a0, false, b1,
                        (short)0, acc[0][1], false, false);
        acc[0][2] = __builtin_amdgcn_wmma_f32_16x16x4_f32(false, a0, false, b2,
                        (short)0, acc[0][2], false, false);
        acc[0][3] = __builtin_amdgcn_wmma_f32_16x16x4_f32(false, a0, false, b3,
                        (short)0, acc[0][3], false, false);
        acc[1][0] = __builtin_amdgcn_wmma_f32_16x16x4_f32(false, a1, false, b0,
                        (short)0, acc[1][0], false, false);
        acc[1][1] = __builtin_amdgcn_wmma_f32_16x16x4_f32(false, a1, false, b1,
                        (short)0, acc[1][1], false, false);
        acc[1][2] = __builtin_amdgcn_wmma_f32_16x16x4_f32(false, a1, false, b2,
                        (short)0, acc[1][2], false, false);
        acc[1][3] = __builtin_amdgcn_wmma_f32_16x16x4_f32(false, a1, false, b3,
                        (short)0, acc[1][3], false, false);
    }

    // C/D layout: VGPR r, lane<16 -> M = r ; lane>=16 -> M = r+8 ; N = lane&15
    const int rowoff = 8 * (lane >> 4);
    const int col    = lane & 15;
#pragma unroll
    for (int mt = 0; mt < 2; ++mt)
#pragma unroll
        for (int j = 0; j < 4; ++j)
#pragma unroll
            for (int r = 0; r < 8; ++r)
                Y[(size_t)(m0 + mt * 16 + rowoff + r) * HID + (n0 + j * 16 + col)]
                    = acc[mt][j][r];
}

// ---------------------------------------------------------------------------
// 6) epilogue: x_out = relu(LN(y + bias) * gamma + beta) [+ x_res]
//    One block (128 thr, 4 waves) per row; float4 per thread.
// ---------------------------------------------------------------------------
__global__ void __launch_bounds__(128) epilogue(const float* __restrict__ Y,
                                                const float* __restrict__ bias,
                                                const float* __restrict__ gam,
                                                const float* __restrict__ bet,
                                                const float* __restrict__ xres,
                                                float* __restrict__ xout,
                                                int addres)
{
    const int row = blockIdx.x;
    const int t   = threadIdx.x;
    const int h   = t * 4;

    const float4 y4 = *(const float4*)&Y[(size_t)row * HID + h];
    const float4 b4 = *(const float4*)&bias[h];
    float vx = y4.x + b4.x, vy = y4.y + b4.y, vz = y4.z + b4.z, vw = y4.w + b4.w;

    float s  = vx + vy + vz + vw;
    float sq = vx * vx + vy * vy + vz * vz + vw * vw;
#pragma unroll
    for (int o = 16; o > 0; o >>= 1) {
        s  += __shfl_down(s,  o, 32);
        sq += __shfl_down(sq, o, 32);
    }
    __shared__ float ssum[4], ssq[4];
    if ((t & 31) == 0) { ssum[t >> 5] = s; ssq[t >> 5] = sq; }
    __syncthreads();
    const float tot  = ssum[0] + ssum[1] + ssum[2] + ssum[3];
    const float tots = ssq[0]  + ssq[1]  + ssq[2]  + ssq[3];
    const float mu   = tot * (1.0f / HID);
    const float var  = tots * (1.0f / HID) - mu * mu;
    const float inv  = rsqrtf(var + 1e-5f);

    const float4 g4 = *(const float4*)&gam[h];
    const float4 e4 = *(const float4*)&bet[h];
    float ox = fmaxf(0.f, (vx - mu) * inv * g4.x + e4.x);
    float oy = fmaxf(0.f, (vy - mu) * inv * g4.y + e4.y);
    float oz = fmaxf(0.f, (vz - mu) * inv * g4.z + e4.z);
    float ow = fmaxf(0.f, (vw - mu) * inv * g4.w + e4.w);
    if (addres) {
        const float4 r4 = *(const float4*)&xres[(size_t)row * HID + h];
        ox += r4.x; oy += r4.y; oz += r4.z; ow += r4.w;
    }
    float4 o; o.x = ox; o.y = oy; o.z = oz; o.w = ow;
    *(float4*)&xout[(size_t)row * HID + h] = o;
}

// ---------------------------------------------------------------------------
// launcher
// ---------------------------------------------------------------------------
extern "C" void kernel_launch(void* const* d_in, const int* in_sizes, int n_in,
                              void* d_out, int out_size, void* d_ws, size_t ws_size,
                              hipStream_t stream)
{
    const float*  node   = (const float*) d_in[0];   // (4,4096,512)
    const float2* coords = (const float2*)d_in[1];   // (4,4096,2)
    const float*  W      = (const float*) d_in[2];   // (3,512,512)
    const float*  bias   = (const float*) d_in[3];   // (3,512)
    const float*  gam    = (const float*) d_in[4];   // (3,512)
    const float*  bet    = (const float*) d_in[5];   // (3,512)
    float* x = (float*)d_out;                        // (4,4096,512), ping-pongs in place

    char* ws = (char*)d_ws;
    int*   knn    = (int*)  (ws + 0);                       //  512 KB
    int*   revcnt = (int*)  (ws + 524288);                  //   64 KB
    int*   rev    = (int*)  (ws + 589824);                  //    4 MB
    float* Wt     = (float*)(ws + 4784128);                 //    3 MB
    float* agg    = (float*)(ws + 7929856);                 // 33.5 MB
    float* y      = (float*)(ws + 41484288);                // 33.5 MB

    knn_kernel <<<NB * NN / 256, 256, 0, stream>>>(coords, knn);
    rev_init   <<<NB * NN / 256, 256, 0, stream>>>(revcnt);
    rev_build  <<<NB * NN * KNN / 256, 256, 0, stream>>>(knn, revcnt, rev);
    transposeW <<<dim3(HID / 16, HID / 16, NLAY), dim3(16, 16), 0, stream>>>(W, Wt);

    for (int l = 0; l < NLAY; ++l) {
        const float* xin = (l == 0) ? node : x;
        aggregate <<<NB * NN, 128, 0, stream>>>(xin, knn, revcnt, rev, agg);
        gemm_wmma <<<dim3(NB * NN / 128, HID / 128), 256, 0, stream>>>(
            agg, Wt + (size_t)l * HID * HID, y);
        epilogue  <<<NB * NN, 128, 0, stream>>>(
            y, bias + l * HID, gam + l * HID, bet + l * HID, x, x, l > 0);
    }
}